// SSM_62861141344380
// MI455X (gfx1250) — compile-verified
//
#include <hip/hip_runtime.h>
#include <math.h>
#include <stdint.h>

// Problem constants (from reference)
#define BSZ 8
#define HD  128
#define LL  8192
#define PP  256
#define CH  256            // scan chunk length (power of 2)
#define LOG2_CH 8
#define NCH (LL / CH)      // 32 chunks

typedef __attribute__((ext_vector_type(2))) float v2f;
typedef __attribute__((ext_vector_type(8))) float v8f;

static __device__ __forceinline__ v8f wmma_f32(v2f a, v2f b, v8f c) {
  // V_WMMA_F32_16X16X4_F32 : D = A(16x4) * B(4x16) + C(16x16)
  return __builtin_amdgcn_wmma_f32_16x16x4_f32(
      /*neg_a=*/false, a, /*neg_b=*/false, b,
      /*c_mod=*/(short)0, c, /*reuse_a=*/false, /*reuse_b=*/false);
}

// CDNA5 async global->LDS copy (16B per lane), tracked by ASYNCcnt.
static __device__ __forceinline__ void async_b128(uint32_t ldsOff, const void* g) {
  asm volatile("global_load_async_to_lds_b128 %0, %1, off"
               :: "v"(ldsOff), "v"(g) : "memory");
}
static __device__ __forceinline__ void wait_async0() {
  asm volatile("s_wait_asynccnt 0x0" ::: "memory");
}
static __device__ __forceinline__ uint32_t lds_off(const void* p) {
  return (uint32_t)(uintptr_t)p;
}

// ---------------------------------------------------------------------------
// Kernel 1: Bu[b][l][p] (complex, split re/im) = u^T(LxH) @ B_bar^T(HxP)
// Wave: 16 p-cols x 4 l-tiles (64 l-rows); B_bar fragments reused 4x,
// 8 independent WMMA accumulators per k-step for ILP.
// ---------------------------------------------------------------------------
__global__ __launch_bounds__(256) void k_gemm_bu(
    const float* __restrict__ u,       // (B,H,L)
    const float* __restrict__ Bre,     // (P,H)
    const float* __restrict__ Bim,     // (P,H)
    float* __restrict__ BuRe,          // (B,L,P)
    float* __restrict__ BuIm) {
  const int wave = (blockIdx.x << 3) + (threadIdx.x >> 5);
  const int lane = threadIdx.x & 31;
  const int pt = wave & 15;                 // 16 p-tiles
  const int lt = (wave >> 4) & 127;         // 128 groups of 64 l
  const int b  = wave >> 11;                // 8 batches
  const int l0 = lt << 6, p0 = pt << 4;
  const int m  = lane & 15;
  const int kb = (lane >> 4) << 1;          // K pair selected by lane half

  v8f accR[4] = {}, accI[4] = {};
  for (int k0 = 0; k0 < HD; k0 += 4) {
    // B fragment: B_bar^T rows = h, cols = p -> element (k,n) = B[p0+n][k0+k]
    v2f br, bi;
    const size_t ba = (size_t)(p0 + m) * HD + k0 + kb;
    br.x = Bre[ba];  br.y = Bre[ba + 1];
    bi.x = Bim[ba];  bi.y = Bim[ba + 1];
    const size_t ua0 = ((size_t)(b * HD + k0 + kb)) * LL + l0 + m;
#pragma unroll
    for (int j = 0; j < 4; ++j) {
      // A fragment: u^T rows = l, cols = h (lane-coalesced in l)
      v2f a;
      a.x = u[ua0 + (j << 4)];
      a.y = u[ua0 + (j << 4) + LL];
      accR[j] = wmma_f32(a, br, accR[j]);
      accI[j] = wmma_f32(a, bi, accI[j]);
    }
  }
  // D layout: VGPR r -> row (l) = +r + 8*(lane>=16), col (p) = p0 + (lane&15)
  const int lofs = (lane >> 4) << 3;
  const int p = p0 + m;
#pragma unroll
  for (int j = 0; j < 4; ++j) {
#pragma unroll
    for (int r = 0; r < 8; ++r) {
      const int l = l0 + (j << 4) + r + lofs;
      const size_t o = ((size_t)b * LL + l) * PP + p;
      BuRe[o] = accR[j][r];
      BuIm[o] = accI[j][r];
    }
  }
}

// ---------------------------------------------------------------------------
// Kernel 2a: per-(b,chunk,p) local scan carry:  x=0; x = lam*(x + Bu_t)
// ---------------------------------------------------------------------------
__global__ __launch_bounds__(256) void k_scan_local(
    const float* __restrict__ BuRe, const float* __restrict__ BuIm,
    const float* __restrict__ Lre,  const float* __restrict__ Lim,
    float* __restrict__ CEr, float* __restrict__ CEi) {
  const int tid = blockIdx.x * 256 + threadIdx.x;
  const int p  = tid & (PP - 1);
  const int ch = (tid >> 8) & (NCH - 1);
  const int b  = tid >> 13;
  const float lr = Lre[p], li = Lim[p];
  float xr = 0.f, xi = 0.f;
  size_t base = ((size_t)b * LL + (size_t)ch * CH) * PP + p;
  for (int k = 0; k < CH; ++k) {
    const float sr = xr + BuRe[base];
    const float si = xi + BuIm[base];
    xr = lr * sr - li * si;
    xi = lr * si + li * sr;
    base += PP;
  }
  const size_t co = ((size_t)(b * NCH + ch)) * PP + p;
  CEr[co] = xr;
  CEi[co] = xi;
}

// ---------------------------------------------------------------------------
// Kernel 2b: combine chunk carries: cin_0=0; cin_{j+1} = lam^CH * cin_j + CE_j
// ---------------------------------------------------------------------------
__global__ __launch_bounds__(256) void k_scan_combine(
    const float* __restrict__ CEr, const float* __restrict__ CEi,
    const float* __restrict__ Lre, const float* __restrict__ Lim,
    float* __restrict__ CIr, float* __restrict__ CIi) {
  const int tid = blockIdx.x * 256 + threadIdx.x;   // B*P threads
  const int p = tid & (PP - 1);
  const int b = tid >> 8;
  // lam^CH via LOG2_CH complex squarings
  float ar = Lre[p], ai = Lim[p];
#pragma unroll
  for (int s = 0; s < LOG2_CH; ++s) {
    const float nr = ar * ar - ai * ai;
    const float ni = 2.f * ar * ai;
    ar = nr; ai = ni;
  }
  float cr = 0.f, ci = 0.f;
  for (int j = 0; j < NCH; ++j) {
    const size_t o = ((size_t)(b * NCH + j)) * PP + p;
    CIr[o] = cr;
    CIi[o] = ci;
    const float nr = ar * cr - ai * ci + CEr[o];
    const float ni = ar * ci + ai * cr + CEi[o];
    cr = nr; ci = ni;
  }
}

// ---------------------------------------------------------------------------
// Kernel 2c: apply: x = carry_in; x = lam*(x + Bu_t); overwrite Bu with x.
// ---------------------------------------------------------------------------
__global__ __launch_bounds__(256) void k_scan_apply(
    float* __restrict__ BuRe, float* __restrict__ BuIm,
    const float* __restrict__ Lre, const float* __restrict__ Lim,
    const float* __restrict__ CIr, const float* __restrict__ CIi) {
  const int tid = blockIdx.x * 256 + threadIdx.x;
  const int p  = tid & (PP - 1);
  const int ch = (tid >> 8) & (NCH - 1);
  const int b  = tid >> 13;
  const float lr = Lre[p], li = Lim[p];
  const size_t co = ((size_t)(b * NCH + ch)) * PP + p;
  float xr = CIr[co], xi = CIi[co];
  size_t base = ((size_t)b * LL + (size_t)ch * CH) * PP + p;
  for (int k = 0; k < CH; ++k) {
    const float sr = xr + BuRe[base];
    const float si = xi + BuIm[base];
    xr = lr * sr - li * si;
    xi = lr * si + li * sr;
    BuRe[base] = xr;
    BuIm[base] = xi;
    base += PP;
  }
}

// ---------------------------------------------------------------------------
// Kernel 3: y[b][o][l] = gelu( Cre@Xre - Cim@Xim + D@u )
// Block = 8 waves, 32-l tile; wave w handles o-tile w x 2 l-subtiles.
// X/u tiles staged via async global->LDS b128 (ASYNCcnt), 16B-aligned pitches,
// bank-conflict-free WMMA B-fragment reads.
// ---------------------------------------------------------------------------
#define K3_LT 2                 // l-subtiles per wave
#define K3_L  (K3_LT * 16)      // 32 l-rows per block
#define XPITCH 260              // floats; 1040 B row (16B multiple), bank stride 4
#define UPITCH 36               // floats; 144 B row (16B multiple)

__global__ __launch_bounds__(256) void k_gemm_out(
    const float* __restrict__ Xre,  // (B,L,P)  = scan output
    const float* __restrict__ Xim,
    const float* __restrict__ u,    // (B,H,L)
    const float* __restrict__ Cre,  // (H,P)
    const float* __restrict__ Cim,  // (H,P)
    const float* __restrict__ Dm,   // (H,H)
    float* __restrict__ y) {        // (B,H,L)
  __shared__ __align__(16) float sXr[K3_L * XPITCH];
  __shared__ __align__(16) float sXi[K3_L * XPITCH];
  __shared__ __align__(16) float sU[HD * UPITCH];

  const int lt = blockIdx.x & (LL / K3_L - 1);   // 256 l-groups
  const int b  = blockIdx.x >> 8;
  const int l0 = lt * K3_L;

  // Stage X tile: K3_L rows (l) x 256 cols (p), stored [l][p] (pitch XPITCH)
  for (int i = threadIdx.x; i < K3_L * (PP / 4); i += 256) {
    const int ll = i >> 6;
    const int q  = (i & 63) << 2;                // p quad
    const size_t g = ((size_t)b * LL + l0 + ll) * PP + q;
    async_b128(lds_off(&sXr[ll * XPITCH + q]), &Xre[g]);
    async_b128(lds_off(&sXi[ll * XPITCH + q]), &Xim[g]);
  }
  // Stage u tile: 128 rows (h) x K3_L cols (l), stored [h][l] (pitch UPITCH)
  for (int i = threadIdx.x; i < HD * (K3_L / 4); i += 256) {
    const int h = i >> 3;
    const int q = (i & 7) << 2;                  // l quad
    async_b128(lds_off(&sU[h * UPITCH + q]),
               &u[((size_t)b * HD + h) * LL + l0 + q]);
  }
  wait_async0();
  __syncthreads();

  const int wv   = threadIdx.x >> 5;
  const int lane = threadIdx.x & 31;
  const int o0 = wv << 4;
  const int m  = lane & 15;
  const int kb = (lane >> 4) << 1;

  v8f acc1[K3_LT] = {};   // Cre@Xre + D@u
  v8f acc2[K3_LT] = {};   // Cim@Xim (subtracted in epilogue)

  // K over P: A rows=o cols=p ; B rows=p cols=l (from LDS)
  for (int k0 = 0; k0 < PP; k0 += 4) {
    v2f car, cai;
    const size_t ca = (size_t)(o0 + m) * PP + k0 + kb;
    car.x = Cre[ca];  car.y = Cre[ca + 1];
    cai.x = Cim[ca];  cai.y = Cim[ca + 1];
#pragma unroll
    for (int j = 0; j < K3_LT; ++j) {
      const int row = (j << 4) + m;
      v2f br, bi;
      br.x = sXr[row * XPITCH + k0 + kb];
      br.y = sXr[row * XPITCH + k0 + kb + 1];
      bi.x = sXi[row * XPITCH + k0 + kb];
      bi.y = sXi[row * XPITCH + k0 + kb + 1];
      acc1[j] = wmma_f32(car, br, acc1[j]);
      acc2[j] = wmma_f32(cai, bi, acc2[j]);
    }
  }
  // K over H: A = D rows=o cols=h ; B = u rows=h cols=l (from LDS)
  for (int k0 = 0; k0 < HD; k0 += 4) {
    v2f da;
    const size_t ia = (size_t)(o0 + m) * HD + k0 + kb;
    da.x = Dm[ia];  da.y = Dm[ia + 1];
#pragma unroll
    for (int j = 0; j < K3_LT; ++j) {
      v2f bu;
      bu.x = sU[(k0 + kb) * UPITCH + (j << 4) + m];
      bu.y = sU[(k0 + kb + 1) * UPITCH + (j << 4) + m];
      acc1[j] = wmma_f32(da, bu, acc1[j]);
    }
  }

  // Epilogue: y = acc1 - acc2, exact-erf GELU; D layout rows=o cols=l
  const int oofs = (lane >> 4) << 3;
#pragma unroll
  for (int j = 0; j < K3_LT; ++j) {
    const int l = l0 + (j << 4) + m;
#pragma unroll
    for (int r = 0; r < 8; ++r) {
      const int o = o0 + r + oofs;
      const float v = acc1[j][r] - acc2[j][r];
      const float g = 0.5f * v * (1.f + erff(v * 0.70710678118654752f));
      y[((size_t)b * HD + o) * LL + l] = g;
    }
  }
}

// ---------------------------------------------------------------------------
extern "C" void kernel_launch(void* const* d_in, const int* in_sizes, int n_in,
                              void* d_out, int out_size, void* d_ws, size_t ws_size,
                              hipStream_t stream) {
  const float* u   = (const float*)d_in[0];   // (B,H,L)
  const float* Lre = (const float*)d_in[1];   // (P,)
  const float* Lim = (const float*)d_in[2];
  const float* Bre = (const float*)d_in[3];   // (P,H)
  const float* Bim = (const float*)d_in[4];
  const float* Cre = (const float*)d_in[5];   // (H,P)
  const float* Cim = (const float*)d_in[6];
  const float* Dm  = (const float*)d_in[7];   // (H,H)
  float* y = (float*)d_out;                   // (B,H,L)

  // Workspace layout
  float* ws = (float*)d_ws;
  const size_t nBu = (size_t)BSZ * LL * PP;            // 16.78M floats each
  float* BuRe = ws;
  float* BuIm = BuRe + nBu;
  const size_t nC = (size_t)BSZ * NCH * PP;            // 65536 floats each
  float* CEr = BuIm + nBu;
  float* CEi = CEr + nC;
  float* CIr = CEi + nC;
  float* CIi = CIr + nC;

  // 1) Bu = B_bar @ u  (complex), stored (B,L,P)
  {
    const int waves = BSZ * (LL / 64) * (PP / 16);     // 16384 waves
    k_gemm_bu<<<waves / 8, 256, 0, stream>>>(u, Bre, Bim, BuRe, BuIm);
  }
  // 2) chunk-parallel complex scan
  {
    const int nthr = BSZ * NCH * PP;                   // 65536
    k_scan_local<<<nthr / 256, 256, 0, stream>>>(BuRe, BuIm, Lre, Lim, CEr, CEi);
    k_scan_combine<<<(BSZ * PP) / 256, 256, 0, stream>>>(CEr, CEi, Lre, Lim, CIr, CIi);
    k_scan_apply<<<nthr / 256, 256, 0, stream>>>(BuRe, BuIm, Lre, Lim, CIr, CIi);
  }
  // 3) y = gelu(Re(C@x) + D@u)
  {
    k_gemm_out<<<BSZ * (LL / K3_L), 256, 0, stream>>>(BuRe, BuIm, u, Cre, Cim, Dm, y);
  }
}